// JaggedLogSoftmaxModule_30150670418629
// MI455X (gfx1250) — compile-verified
//
#include <hip/hip_runtime.h>
#include <math.h>

// Jagged log-softmax, MI455X (gfx1250).
//
// One wave32 per segment, 8 segments per 256-thread block, no workgroup
// barriers (each wave waits only on its own ASYNCcnt). Segment is staged
// HBM -> LDS with global_load_async_to_lds_b128 (16B/lane, 512B/wave-instr),
// with a <=3-element scalar prologue/tail to fix up 4B-aligned segment
// starts. LDS layout is skewed by (beg & 3) elements so the 16B-aligned
// global chunks are also 16B-aligned in LDS. All three passes (max,
// sum-exp, write) run on float4 chunks: ds_load_b128 + global_store_b128.
// HBM traffic: exactly one read + one write per element (~257MB -> ~11us).

#define SEG_PER_BLOCK 8
#define LDS_CAP       1024             // max segment length held in LDS (~772 expected)
#define LDS_STRIDE    1028             // floats per wave slice (1028*4 = 4112 B, 16B multiple)

__device__ __forceinline__ float wave_max32(float v) {
    #pragma unroll
    for (int o = 16; o > 0; o >>= 1)
        v = fmaxf(v, __shfl_xor(v, o, 32));
    return v;
}

__device__ __forceinline__ float wave_sum32(float v) {
    #pragma unroll
    for (int o = 16; o > 0; o >>= 1)
        v += __shfl_xor(v, o, 32);
    return v;
}

__global__ __launch_bounds__(256) void jagged_logsoftmax_kernel(
    const float* __restrict__ logits,
    const long long* __restrict__ prefix,   // int64 exclusive segment ends
    float* __restrict__ out,
    int nseg)
{
    __shared__ __align__(16) float smem[SEG_PER_BLOCK * LDS_STRIDE];

    const int wv   = threadIdx.x >> 5;
    const int lane = threadIdx.x & 31;
    const int seg  = blockIdx.x * SEG_PER_BLOCK + wv;
    if (seg >= nseg) return;

    const long long beg = (seg == 0) ? 0ll : prefix[seg - 1];
    const long long end = prefix[seg];
    const int len = (int)(end - beg);
    if (len <= 0) return;

    const float* __restrict__ src = logits + beg;
    float*       __restrict__ dst = out    + beg;

    float m = -__builtin_inff();
    float s = 0.0f;

    if (len <= LDS_CAP) {
        const int ofs = (int)(beg & 3);          // element misalignment of src/dst
        int p = (4 - ofs) & 3;                   // scalar prologue count
        if (p > len) p = len;
        const int nvec = (len - p) >> 2;         // float4 chunks
        const int tail = p + (nvec << 2);        // first scalar tail index

        float* buf = &smem[wv * LDS_STRIDE];     // element i lives at buf[ofs + i]
        const unsigned lds0 = (unsigned)(uintptr_t)buf;

        // ---- Stage HBM -> LDS via CDNA5 async loads (ASYNCcnt) ----
        if (lane < p) {                          // prologue (<=3 elements)
            unsigned la = lds0 + 4u * (unsigned)(ofs + lane);
            unsigned long long ga = (unsigned long long)(uintptr_t)(src + lane);
            asm volatile("global_load_async_to_lds_b32 %0, %1, off"
                         :: "v"(la), "v"(ga) : "memory");
        }
        for (int c = lane; c < nvec; c += 32) {  // 16B-aligned bulk
            int i = p + (c << 2);
            unsigned la = lds0 + 4u * (unsigned)(ofs + i);
            unsigned long long ga = (unsigned long long)(uintptr_t)(src + i);
            asm volatile("global_load_async_to_lds_b128 %0, %1, off"
                         :: "v"(la), "v"(ga) : "memory");
        }
        for (int i = tail + lane; i < len; i += 32) {   // tail (<=3 elements)
            unsigned la = lds0 + 4u * (unsigned)(ofs + i);
            unsigned long long ga = (unsigned long long)(uintptr_t)(src + i);
            asm volatile("global_load_async_to_lds_b32 %0, %1, off"
                         :: "v"(la), "v"(ga) : "memory");
        }
        // Buffer is wave-private: wait on this wave's ASYNCcnt only.
        asm volatile("s_wait_asynccnt 0" ::: "memory");

        const float4* bufv = (const float4*)(buf + ofs + p);   // 16B-aligned
        float4*       dstv = (float4*)(dst + p);               // 16B-aligned

        // ---- Pass 1: max ----
        if (lane < p)                      m = fmaxf(m, buf[ofs + lane]);
        for (int i = tail + lane; i < len; i += 32) m = fmaxf(m, buf[ofs + i]);
        for (int c = lane; c < nvec; c += 32) {
            float4 x = bufv[c];            // ds_load_b128
            m = fmaxf(fmaxf(m, fmaxf(x.x, x.y)), fmaxf(x.z, x.w));
        }
        m = wave_max32(m);

        // ---- Pass 2: sum exp(x - m) ----
        if (lane < p)                      s += __expf(buf[ofs + lane] - m);
        for (int i = tail + lane; i < len; i += 32) s += __expf(buf[ofs + i] - m);
        for (int c = lane; c < nvec; c += 32) {
            float4 x = bufv[c];
            s += __expf(x.x - m) + __expf(x.y - m) + __expf(x.z - m) + __expf(x.w - m);
        }
        s = wave_sum32(s);

        const float cst = m + __logf(s);

        // ---- Pass 3: write x - m - log(sum) ----
        if (lane < p)                      dst[lane] = buf[ofs + lane] - cst;
        for (int i = tail + lane; i < len; i += 32) dst[i] = buf[ofs + i] - cst;
        for (int c = lane; c < nvec; c += 32) {
            float4 x = bufv[c];
            dstv[c] = make_float4(x.x - cst, x.y - cst, x.z - cst, x.w - cst);  // global_store_b128
        }
    } else {
        // Oversized-segment fallback (not expected with this generator):
        // stream from global; re-reads hit L2.
        for (int i = lane; i < len; i += 32) m = fmaxf(m, src[i]);
        m = wave_max32(m);
        for (int i = lane; i < len; i += 32) s += __expf(src[i] - m);
        s = wave_sum32(s);
        const float cst = m + __logf(s);
        for (int i = lane; i < len; i += 32) dst[i] = src[i] - cst;
    }
}

extern "C" void kernel_launch(void* const* d_in, const int* in_sizes, int n_in,
                              void* d_out, int out_size, void* d_ws, size_t ws_size,
                              hipStream_t stream) {
    const float*     logits = (const float*)d_in[0];
    const long long* prefix = (const long long*)d_in[1];
    float* out = (float*)d_out;
    const int nseg = in_sizes[1];

    const int blocks = (nseg + SEG_PER_BLOCK - 1) / SEG_PER_BLOCK;
    hipLaunchKernelGGL(jagged_logsoftmax_kernel,
                       dim3(blocks), dim3(256), 0, stream,
                       logits, prefix, out, nseg);
}